// PrunedRNNTLoss_30030411334201
// MI455X (gfx1250) — compile-verified
//
#include <hip/hip_runtime.h>

// Pruned RNNT loss, banded DP, B=8 T=200 U=200 V=512 (V hardcoded per reference).
// Strategy: async-gather the 11-wide band (blank + target-label log-probs) into
// LDS via CDNA5 GLOBAL_LOAD_ASYNC_TO_LDS_B32, then run the T-step band DP on
// one wave32 using shuffle-based prefix-sum / cum-logsumexp scans.

#define PRUNE 5
#define BANDW 11            // band width = 2*PRUNE+1
#define NEGV  (-1e30f)
#define MAXT  256

__global__ __launch_bounds__(256)
void rnnt_band_kernel(const float* __restrict__ lp,
                      const int*   __restrict__ targets,
                      const int*   __restrict__ logit_lengths,
                      const int*   __restrict__ target_lengths,
                      float*       __restrict__ ws,
                      int T, int U, int V) {
  if (T > MAXT) return;
  const int b = blockIdx.x;

  // Band-relative LDS arrays: slot (t, j) <-> u = t - PRUNE + j
  __shared__ float Pb[MAXT * BANDW];   // blank log-prob  pb[t][u]
  __shared__ float Lb[MAXT * BANDW];   // label log-prob  L[t][u] = lp[t,u,tgt[u]]

  const uint32_t pbOff = (uint32_t)(uintptr_t)&Pb[0];  // LDS byte offsets
  const uint32_t lbOff = (uint32_t)(uintptr_t)&Lb[0];

  const int S = T * BANDW;
  const int*   tgt = targets + b * U;
  const float* lpb = lp + (size_t)b * T * U * V;

  // ---- Phase 1: async gather band values straight into LDS ----
  for (int s = threadIdx.x; s < S; s += blockDim.x) {
    int t = s / BANDW;
    int j = s - t * BANDW;
    int u = t - PRUNE + j;
    if (u >= 0 && u < U) {
      const float* gp = lpb + ((size_t)t * U + u) * V;     // lp[b,t,u,0]
      uint32_t dp = pbOff + (uint32_t)s * 4u;
      asm volatile("global_load_async_to_lds_b32 %0, %1, off"
                   :: "v"(dp), "v"(gp) : "memory");
      const float* gl = gp + tgt[u];                       // lp[b,t,u,tgt[u]]
      uint32_t dl = lbOff + (uint32_t)s * 4u;
      asm volatile("global_load_async_to_lds_b32 %0, %1, off"
                   :: "v"(dl), "v"(gl) : "memory");
    }
  }
  asm volatile("s_wait_asynccnt 0" ::: "memory");
  __syncthreads();

  // ---- Phase 2: banded DP on wave 0 (lane j = band slot, u = t-PRUNE+lane) ----
  if (threadIdx.x < 32) {
    const int lane = threadIdx.x;
    const int Tb = logit_lengths[b];
    const int Ub = target_lengths[b];

    float alpha;
    // t = 0: alpha0[u] = sum_{k<u} L[0,k] for u <= PRUNE (band start at u=0)
    {
      int u = 0 - PRUNE + lane;
      bool valid = (lane < BANDW) && (u >= 0) && (u < U);
      float Lv = valid ? Lb[lane] : 0.0f;
      float inc = Lv;
      #pragma unroll
      for (int off = 1; off < 16; off <<= 1) {
        float y = __shfl_up(inc, off, 32);
        if (lane >= off) inc += y;
      }
      float e = __shfl_up(inc, 1, 32);
      if (lane == 0) e = 0.0f;
      alpha = valid ? e : NEGV;
    }

    for (int t = 1; t <= Tb; ++t) {
      int u = t - PRUNE + lane;
      bool valid = (lane < BANDW) && (u >= 0) && (u < U);

      // prev alpha at same u: band of t-1 shifts by one slot -> shuffle down.
      // Lanes >= 11 hold NEGV, so lane 10 correctly pulls NEGV (outside prev band).
      float ap  = __shfl_down(alpha, 1, 32);
      float pbr = (valid && lane < BANDW - 1) ? Pb[(t - 1) * BANDW + (lane + 1)] : 0.0f;
      float A   = valid ? (ap + pbr) : NEGV;

      // Band-local emission prefix sum e_j (global ecum constant cancels exactly).
      float Lv = valid ? Lb[t * BANDW + lane] : 0.0f;
      float inc = Lv;
      #pragma unroll
      for (int off = 1; off < 16; off <<= 1) {
        float y = __shfl_up(inc, off, 32);
        if (lane >= off) inc += y;
      }
      float e = __shfl_up(inc, 1, 32);
      if (lane == 0) e = 0.0f;

      // Inclusive cumulative log-sum-exp of (A - e) across band lanes.
      float x = A - e;
      #pragma unroll
      for (int off = 1; off < 16; off <<= 1) {
        float y = __shfl_up(x, off, 32);
        if (lane >= off) {
          float m = fmaxf(x, y);
          x = m + __logf(__expf(x - m) + __expf(y - m));
        }
      }
      alpha = valid ? (e + x) : NEGV;
    }

    // loss_b = -alpha[Tb, Ub]; band slot of Ub at t=Tb is Ub - Tb + PRUNE (in [0,5])
    if (lane == Ub - Tb + PRUNE) ws[b] = -alpha;
  }
}

__global__ void rnnt_mean_kernel(const float* __restrict__ ws,
                                 float* __restrict__ out, int B) {
  if (threadIdx.x == 0) {
    float s = 0.0f;
    for (int i = 0; i < B; ++i) s += ws[i];
    out[0] = s / (float)B;
  }
}

extern "C" void kernel_launch(void* const* d_in, const int* in_sizes, int n_in,
                              void* d_out, int out_size, void* d_ws, size_t ws_size,
                              hipStream_t stream) {
  const float* lp      = (const float*)d_in[0];
  const int*   targets = (const int*)  d_in[1];
  const int*   ll      = (const int*)  d_in[2];
  const int*   tl      = (const int*)  d_in[3];

  const int B = in_sizes[2];
  const int U = in_sizes[1] / B;
  const int V = 512;  // fixed by reference problem
  const int T = (int)((long long)in_sizes[0] / ((long long)B * U * V));

  float* ws = (float*)d_ws;
  rnnt_band_kernel<<<B, 256, 0, stream>>>(lp, targets, ll, tl, ws, T, U, V);
  rnnt_mean_kernel<<<1, 32, 0, stream>>>(ws, (float*)d_out, B);
}